// CLI_63702954934481
// MI455X (gfx1250) — compile-verified
//
#include <hip/hip_runtime.h>

typedef __attribute__((ext_vector_type(2))) float v2f;
typedef __attribute__((ext_vector_type(8))) float v8f;

#define B_SZ 8
#define NA 4096
#define NB 4096
#define FD 64
#define WAVES 8
#define ROWS_PER_WAVE 16
#define ROWS_PER_BLOCK (WAVES * ROWS_PER_WAVE)
#define NTILES (NB / 16)
#define KBIAS 48387              // 3*127*127: makes key = 2*cross - nb2 + KBIAS >= 0

// Branchless insert of pk into sorted-descending {b0,b1,b2}: pure min/max network.
#define TOP3_INSERT_PK(pk, b0, b1, b2) do {          \
    int _r0 = min((pk), (b0));                        \
    (b0)    = max((pk), (b0));                        \
    int _r1 = min(_r0, (b1));                         \
    (b1)    = max(_r0, (b1));                         \
    (b2)    = max(_r1, (b2));                         \
  } while (0)

__global__ __launch_bounds__(256) void knn3_wmma(
    const float* __restrict__ a_feats,
    const float* __restrict__ b_feats,
    const int*   __restrict__ a_coords,
    const int*   __restrict__ b_coords,
    float*       __restrict__ out)
{
  __shared__ unsigned sPack[NB];                 // x | y<<8 | z<<16 (0..127 each)
  __shared__ int      sKb2[NB];                  // ((KBIAS-|cb|^2)<<12) + (4095-n)
  __shared__ int      sPk[WAVES][ROWS_PER_WAVE][3];

  const int tid   = threadIdx.x;
  const int lane  = tid & 31;
  const int wave  = tid >> 5;
  const int mlane = lane & 15;
  const bool hi   = lane >= 16;

  const int blocksPerBatch = NA / ROWS_PER_BLOCK;
  const int b       = blockIdx.x / blocksPerBatch;
  const int rowBase = (blockIdx.x % blocksPerBatch) * ROWS_PER_BLOCK
                    + wave * ROWS_PER_WAVE;

  // ---- Stage 1: block-cooperative preload of quantized b coords into LDS ----
  for (int n = tid; n < NB; n += 256) {
    const int* p = b_coords + ((size_t)b * NB + n) * 3;
    int x = p[0] >> 4, y = p[1] >> 4, z = p[2] >> 4;
    sPack[n] = (unsigned)x | ((unsigned)y << 8) | ((unsigned)z << 16);
    sKb2[n]  = ((KBIAS - (x * x + y * y + z * z)) << 12) + (4095 - n);
  }
  __syncthreads();

  // ---- Stage 2: A operand = 2*ca (16x4 f32; K = {2x, 2y, 2z, 0}) ----
  // Doubling A makes the WMMA emit 2*cross directly (still exact in f32).
  // Layout: lanes 0-15 hold (K0,K1) of row M=lane; lanes 16-31 hold (K2,K3).
  const int arow = rowBase + mlane;
  const int* ap = a_coords + ((size_t)b * NA + arow) * 3;
  const int iax = ap[0] >> 4, iay = ap[1] >> 4, iaz = ap[2] >> 4;
  const int na2i = iax * iax + iay * iay + iaz * iaz;
  v2f A;
  A.x = hi ? (float)(2 * iaz) : (float)(2 * iax);
  A.y = hi ? 0.0f             : (float)(2 * iay);

  int bp0[8], bp1[8], bp2[8];
#pragma unroll
  for (int r = 0; r < 8; ++r) { bp0[r] = bp1[r] = bp2[r] = -1; }

  const v8f Cz = {};   // zero accumulator -> inline-0 C operand

  // ---- Stage 3: sweep all Nb columns 16 at a time ----
  // Dv[r] = 2*(ca_m . cb_n);  pk = (ik<<12) + kb2 = (key<<12) + (4095-n).
  // max pk  <=>  min distance, ties broken toward lower index (as jax top_k).
  for (int t = 0; t < NTILES; ++t) {
    const int n = t * 16 + mlane;
    const unsigned u = sPack[n];
    const int kb2 = sKb2[n];
    const float bx = (float)(u & 255u);
    const float by = (float)((u >> 8) & 255u);
    const float bz = (float)((u >> 16) & 255u);
    v2f Bm;                       // B 4x16: lanes 0-15 hold K0,K1; 16-31 hold K2,K3
    Bm.x = hi ? bz : bx;
    Bm.y = hi ? 0.0f : by;

    v8f Dv = __builtin_amdgcn_wmma_f32_16x16x4_f32(
        /*neg_a=*/false, A, /*neg_b=*/false, Bm,
        /*c_mod=*/(short)0, Cz, /*reuse_a=*/false, /*reuse_b=*/false);

#pragma unroll
    for (int r = 0; r < 8; ++r) {
      const int ik = (int)Dv[r];           // exact: 2*cross, integer
      const int pk = (ik << 12) + kb2;     // single v_lshl_add_u32
      TOP3_INSERT_PK(pk, bp0[r], bp1[r], bp2[r]);
    }
  }

  // ---- Stage 4: butterfly merge across the 16 lanes of each half-wave ----
#pragma unroll
  for (int r = 0; r < 8; ++r) {
    int p0 = bp0[r], p1 = bp1[r], p2 = bp2[r];
#pragma unroll
    for (int off = 1; off < 16; off <<= 1) {
      int q0 = __shfl_xor(p0, off, 32);
      int q1 = __shfl_xor(p1, off, 32);
      int q2 = __shfl_xor(p2, off, 32);
      TOP3_INSERT_PK(q0, p0, p1, p2);
      TOP3_INSERT_PK(q1, p0, p1, p2);
      TOP3_INSERT_PK(q2, p0, p1, p2);
    }
    if (mlane == 0) {
      const int rowLocal = r + (hi ? 8 : 0);
      sPk[wave][rowLocal][0] = p0;
      sPk[wave][rowLocal][1] = p1;
      sPk[wave][rowLocal][2] = p2;
    }
  }
  __syncthreads();

  // ---- Stage 5: unpack, weights, feature gather, coalesced 128-float rows ----
  const float* afBase = a_feats + ((size_t)b * NA + rowBase) * FD;
  float* outBase = out + ((size_t)b * NA + rowBase) * (2 * FD);
  const float* bfB = b_feats + (size_t)b * NB * FD;

  for (int row = 0; row < ROWS_PER_WAVE; ++row) {
    const int pk0 = sPk[wave][row][0];
    const int pk1 = sPk[wave][row][1];
    const int pk2 = sPk[wave][row][2];
    const int rna2 = __shfl(na2i, row, 32);      // |ca_row|^2 lives in lane==row
    const int j0 = 4095 - (pk0 & 4095);
    const int j1 = 4095 - (pk1 & 4095);
    const int j2 = 4095 - (pk2 & 4095);
    // d2 = na2 + nb2 - 2*cross = na2 + KBIAS - key   (exact integer, >= 0)
    const int d20 = rna2 + KBIAS - (int)((unsigned)pk0 >> 12);
    const int d21 = rna2 + KBIAS - (int)((unsigned)pk1 >> 12);
    const int d22 = rna2 + KBIAS - (int)((unsigned)pk2 >> 12);
    const float inv = 1.0f / 128.0f;
    const float w0 = fmaxf(0.5f - sqrtf((float)d20) * inv, 0.0f);
    const float w1 = fmaxf(0.5f - sqrtf((float)d21) * inv, 0.0f);
    const float w2 = fmaxf(0.5f - sqrtf((float)d22) * inv, 0.0f);
    const float* bf0 = bfB + (size_t)j0 * FD;
    const float* bf1 = bfB + (size_t)j1 * FD;
    const float* bf2 = bfB + (size_t)j2 * FD;
    const float* af  = afBase + (size_t)row * FD;
    float* op = outBase + (size_t)row * (2 * FD);
#pragma unroll
    for (int h = 0; h < 2; ++h) {
      const int d = lane + h * 32;
      const float tmp = w0 * bf0[d] + w1 * bf1[d] + w2 * bf2[d];
      op[d]      = af[d];
      op[FD + d] = tmp;
    }
  }
}

extern "C" void kernel_launch(void* const* d_in, const int* in_sizes, int n_in,
                              void* d_out, int out_size, void* d_ws, size_t ws_size,
                              hipStream_t stream) {
  const float* a_feats  = (const float*)d_in[0];
  const float* b_feats  = (const float*)d_in[1];
  const int*   a_coords = (const int*)d_in[2];
  const int*   b_coords = (const int*)d_in[3];
  float* out = (float*)d_out;

  dim3 grid(B_SZ * (NA / ROWS_PER_BLOCK));   // 8 * 32 = 256 blocks
  dim3 block(256);                           // 8 wave32 per block
  knn3_wmma<<<grid, block, 0, stream>>>(a_feats, b_feats, a_coords, b_coords, out);
}